// KalmanFilter_81088982548894
// MI455X (gfx1250) — compile-verified
//
#include <hip/hip_runtime.h>

// CDNA5 (gfx1250, wave32) Kalman filter: one wave per group, P kept in the
// V_WMMA_F32_16X16X4_F32 C/D register layout, all 16x16 matmuls done as
// chains of K=4 f32 WMMAs; LDS staging for C/D <-> A/B operand layout swaps.

typedef __attribute__((ext_vector_type(2))) float v2f;
typedef __attribute__((ext_vector_type(8))) float v8f;

namespace {
constexpr int kG = 1024, kT = 200, kO = 8, kS = 16;
constexpr int WPB = 4;  // waves per block
// per-wave LDS floats: MAT(256) MAT2(256) GJ(128) mv(16) mv2(16) yv(8) Hs(128) Fs(256)
constexpr int LDSW = 256 + 256 + 128 + 16 + 16 + 8 + 128 + 256;
}

__device__ __forceinline__ v8f wmma_f32(v2f a, v2f b, v8f c) {
  // D = A(16x4) * B(4x16) + C ; f32 WMMA forbids A/B negation (NEG[1:0]=0)
  return __builtin_amdgcn_wmma_f32_16x16x4_f32(false, a, false, b, (short)0, c,
                                               false, false);
}

// A-operand layout, chunk c covers K=4c..4c+3:
//   VGPR 2c  : lanes0-15 K=4c,   lanes16-31 K=4c+2   (M = lane%16)
//   VGPR 2c+1: lanes0-15 K=4c+1, lanes16-31 K=4c+3
__device__ __forceinline__ v2f ldA(const float* M, int c, int lh, int ll) {
  int k = 4 * c + 2 * lh;
  v2f a;
  a.x = M[ll * 16 + k];
  a.y = M[ll * 16 + k + 1];
  return a;
}
// B-operand layout (mirror of A under M<->N): element (K, N=lane%16)
__device__ __forceinline__ v2f ldB(const float* M, int c, int lh, int ll) {
  int k = 4 * c + 2 * lh;
  v2f b;
  b.x = M[k * 16 + ll];
  b.y = M[(k + 1) * 16 + ll];
  return b;
}
// C/D layout: VGPR r holds (M = r + 8*(lane/16), N = lane%16)
__device__ __forceinline__ void stCD(float* M, v8f d, int lh, int ll) {
#pragma unroll
  for (int r = 0; r < 8; ++r) M[(r + 8 * lh) * 16 + ll] = d[r];
}

__global__ void __launch_bounds__(32 * WPB)
kalman_kernel(const float* __restrict__ obs,  // [G,T,O]
              const float* __restrict__ Fg,   // [S,S]
              const float* __restrict__ Hg,   // [O,S]
              const float* __restrict__ Qg,   // [S,S]
              const float* __restrict__ Rg,   // [O,O]
              float* __restrict__ out)        // means [T,G,S] ++ covs [T,G,S,S]
{
  __shared__ float lds[WPB * LDSW];
  const int lane = threadIdx.x & 31;
  const int wave = threadIdx.x >> 5;
  const int g = blockIdx.x * WPB + wave;
  const int lh = lane >> 4, ll = lane & 15;

  float* MAT  = lds + wave * LDSW;  // staging tile #1 (P, then K)
  float* MAT2 = MAT + 256;          // staging tile #2 (PHt, HP, Pu, T1)
  float* GJ   = MAT2 + 256;         // 8x16 Gauss-Jordan workspace
  float* mv   = GJ + 128;           // mean (persistent)
  float* mv2  = mv + 16;            // updated mean
  float* yv   = mv2 + 16;           // innovation residual
  float* Hs   = yv + 8;             // H copy for mat-vec dots
  float* Fs   = Hs + 128;           // F copy for mat-vec dots

  for (int e = lane; e < kO * kS; e += 32) Hs[e] = Hg[e];
  for (int e = lane; e < kS * kS; e += 32) Fs[e] = Fg[e];

  // Constants in operand layouts. Note HB serves BOTH as A-operand for H
  // (A(M=ll,K)=H[ll][K], rows>=8 zero) and as B-operand for H^T
  // (B(K,N=ll)=H[ll][K], cols>=8 zero) — same per-lane values. Same for FB.
  v2f HB[4], FB[4];
  v8f Qcd, Rcd;
#pragma unroll
  for (int c = 0; c < 4; ++c) {
    int k = 4 * c + 2 * lh;
    HB[c].x = (ll < kO) ? Hg[ll * kS + k] : 0.f;
    HB[c].y = (ll < kO) ? Hg[ll * kS + k + 1] : 0.f;
    FB[c].x = Fg[ll * kS + k];
    FB[c].y = Fg[ll * kS + k + 1];
  }
#pragma unroll
  for (int r = 0; r < 8; ++r) {
    int m = r + 8 * lh;
    Qcd[r] = Qg[m * kS + ll];
    Rcd[r] = (m < kO && ll < kO) ? Rg[m * kO + ll] : 0.f;  // R padded into 16x16
  }

  // initial state: m = 0, P = I (in C/D layout)
  v8f Pcd;
#pragma unroll
  for (int r = 0; r < 8; ++r) Pcd[r] = ((r + 8 * lh) == ll) ? 1.f : 0.f;
  if (lane < kS) mv[lane] = 0.f;

  float* means = out;
  float* covs = out + (size_t)kT * kG * kS;

  for (int t = 0; t < kT; ++t) {
    // ---- stage P; fetch as A (for P*H^T) and B (for H*P) operands
    stCD(MAT, Pcd, lh, ll);
    v2f PA[4], PB[4];
#pragma unroll
    for (int c = 0; c < 4; ++c) { PA[c] = ldA(MAT, c, lh, ll); PB[c] = ldB(MAT, c, lh, ll); }

    v8f PHt = {}, HP = {};
#pragma unroll
    for (int c = 0; c < 4; ++c) PHt = wmma_f32(PA[c], HB[c], PHt);  // P*H^T (cols>=8 zero)
#pragma unroll
    for (int c = 0; c < 4; ++c) HP = wmma_f32(HB[c], PB[c], HP);    // H*P  (rows>=8 zero)

    // ---- stage PHt; need B layout (for H*PHt) and A chunks 0..1 (for K)
    stCD(MAT2, PHt, lh, ll);
    v2f PHtB[4], PHtA[2];
#pragma unroll
    for (int c = 0; c < 4; ++c) PHtB[c] = ldB(MAT2, c, lh, ll);
#pragma unroll
    for (int c = 0; c < 2; ++c) PHtA[c] = ldA(MAT2, c, lh, ll);

    // ---- S = H*PHt + R (nonzero only in top-left 8x8)
    v8f Scd = Rcd;
#pragma unroll
    for (int c = 0; c < 4; ++c) Scd = wmma_f32(HB[c], PHtB[c], Scd);

    // ---- invert 8x8 S: lane-parallel Gauss-Jordan on [S | I] in LDS.
    if (lane < 8) {
#pragma unroll
      for (int r = 0; r < 8; ++r) GJ[r * 16 + lane] = Scd[r];  // column `lane`
    } else if (lane < 16) {
      int cc = lane - 8;
#pragma unroll
      for (int r = 0; r < 8; ++r) GJ[r * 16 + 8 + cc] = (r == cc) ? 1.f : 0.f;
    }
    for (int p = 0; p < 8; ++p) {
      float pivinv = 1.0f / GJ[p * 16 + p];
      if (lane < 16) GJ[p * 16 + lane] *= pivinv;  // scale pivot row
      float fr[4], gv[4], ov[4];
      int rw[4], cl[4];
#pragma unroll
      for (int i = 0; i < 4; ++i) {  // 128 entries / 32 lanes: all loads first
        int e = i * 32 + lane;
        rw[i] = e >> 4;
        cl[i] = e & 15;
        fr[i] = (rw[i] == p) ? 0.f : GJ[rw[i] * 16 + p];
        gv[i] = GJ[p * 16 + cl[i]];
        ov[i] = GJ[rw[i] * 16 + cl[i]];
      }
#pragma unroll
      for (int i = 0; i < 4; ++i) GJ[rw[i] * 16 + cl[i]] = ov[i] - fr[i] * gv[i];
    }

    // Sinv (GJ cols 8..15) as B operand; only chunks 0..1 (K<8) are nonzero
    v2f SiB[2];
#pragma unroll
    for (int c = 0; c < 2; ++c) {
      int k = 4 * c + 2 * lh;
      SiB[c].x = (ll < kO) ? GJ[k * 16 + 8 + ll] : 0.f;
      SiB[c].y = (ll < kO) ? GJ[(k + 1) * 16 + 8 + ll] : 0.f;
    }

    // ---- K = PHt * Sinv (contraction over o<8 -> 2 WMMAs)
    v8f Kcd = {};
    Kcd = wmma_f32(PHtA[0], SiB[0], Kcd);
    Kcd = wmma_f32(PHtA[1], SiB[1], Kcd);

    // stage K (also read by the mean update); fetch -K as A operand
    stCD(MAT, Kcd, lh, ll);
    v2f nKA[2];
#pragma unroll
    for (int c = 0; c < 2; ++c) {
      v2f a = ldA(MAT, c, lh, ll);
      nKA[c].x = -a.x;
      nKA[c].y = -a.y;
    }

    // stage HP; rows>=8 zero so only B chunks 0..1 contribute
    stCD(MAT2, HP, lh, ll);
    v2f HPB[2];
#pragma unroll
    for (int c = 0; c < 2; ++c) HPB[c] = ldB(MAT2, c, lh, ll);

    // ---- P_u = P + (-K)*(H P)   (f32 WMMA has no A-neg, so negate K in VALU)
    v8f Pu = Pcd;
    Pu = wmma_f32(nKA[0], HPB[0], Pu);
    Pu = wmma_f32(nKA[1], HPB[1], Pu);

    // ---- mean path: y = obs - H m ; m_u = m + K y ; m = F m_u
    if (lane < kO) {
      float acc = obs[(size_t)g * kT * kO + (size_t)t * kO + lane];
      for (int s = 0; s < kS; ++s) acc -= Hs[lane * kS + s] * mv[s];
      yv[lane] = acc;
    }
    if (lane < kS) {
      float acc = mv[lane];
      for (int o = 0; o < kO; ++o) acc += MAT[lane * 16 + o] * yv[o];
      mv2[lane] = acc;
    }
    if (lane < kS) {
      float acc = 0.f;
      for (int j = 0; j < kS; ++j) acc += Fs[lane * kS + j] * mv2[j];
      mv[lane] = acc;
      __builtin_nontemporal_store(acc, &means[(size_t)t * kG * kS + (size_t)g * kS + lane]);
    }

    // ---- P = F * Pu * F^T + Q
    stCD(MAT2, Pu, lh, ll);
    v2f PuB[4];
#pragma unroll
    for (int c = 0; c < 4; ++c) PuB[c] = ldB(MAT2, c, lh, ll);
    v8f T1 = {};
#pragma unroll
    for (int c = 0; c < 4; ++c) T1 = wmma_f32(FB[c], PuB[c], T1);  // F * Pu

    stCD(MAT2, T1, lh, ll);
    v2f T1A[4];
#pragma unroll
    for (int c = 0; c < 4; ++c) T1A[c] = ldA(MAT2, c, lh, ll);
    Pcd = Qcd;
#pragma unroll
    for (int c = 0; c < 4; ++c) Pcd = wmma_f32(T1A[c], FB[c], Pcd);  // T1 * F^T + Q

    // ---- stream out covariance (non-temporal: 210 MB, don't thrash L2)
    float* dst = covs + (size_t)t * kG * kS * kS + (size_t)g * kS * kS;
#pragma unroll
    for (int r = 0; r < 8; ++r)
      __builtin_nontemporal_store(Pcd[r], &dst[(r + 8 * lh) * kS + ll]);
  }
}

extern "C" void kernel_launch(void* const* d_in, const int* in_sizes, int n_in,
                              void* d_out, int out_size, void* d_ws, size_t ws_size,
                              hipStream_t stream) {
  (void)in_sizes; (void)n_in; (void)out_size; (void)d_ws; (void)ws_size;
  const float* obs = (const float*)d_in[0];
  const float* F = (const float*)d_in[1];
  const float* H = (const float*)d_in[2];
  const float* Q = (const float*)d_in[3];
  const float* R = (const float*)d_in[4];
  float* out = (float*)d_out;
  dim3 grid(kG / WPB), block(32 * WPB);
  hipLaunchKernelGGL(kalman_kernel, grid, block, 0, stream, obs, F, H, Q, R, out);
}